// circuit_16750372454455
// MI455X (gfx1250) — compile-verified
//
#include <hip/hip_runtime.h>
#include <math.h>

#define NQB 23
#define NSTATE (1u << NQB)

typedef __attribute__((ext_vector_type(2))) float v2f;
typedef __attribute__((ext_vector_type(8))) float v8f;

// ---------------------------------------------------------------------------
// K0: build 2x2 gate matrices and the fused 16x16 complex matrix for the
// low flat-bit nibble (qubits 19..22). Matrix stored transposed [k*16+n]
// as Re, Im, and -Im (f32 WMMA cannot negate A/B operands).
// ---------------------------------------------------------------------------
__global__ __launch_bounds__(256) void qc_prepare(
    const float* __restrict__ param,
    float* __restrict__ gates,   // 23*8 floats: U[q][i][j] (re,im)
    float* __restrict__ mret,    // 256: Re M[n][k] at [k*16+n]
    float* __restrict__ mimt,    // 256: Im
    float* __restrict__ mngt)    // 256: -Im
{
  __shared__ float g[23 * 8];
  const int t = threadIdx.x;
  if (t < 23) {
    float a = param[3 * t + 0];
    float b = param[3 * t + 1];
    float c = param[3 * t + 2];
    float cb = cosf(0.5f * b), sb = sinf(0.5f * b);
    float apc = 0.5f * (a + c), amc = 0.5f * (a - c);
    float capc = cosf(apc), sapc = sinf(apc);
    float camc = cosf(amc), samc = sinf(amc);
    float* gq = g + t * 8;
    gq[0] =  cb * capc; gq[1] = -cb * sapc;   // U00 = cb * e^{-i(a+c)/2}
    gq[2] = -sb * camc; gq[3] =  sb * samc;   // U01 = -sb * e^{-i(a-c)/2}
    gq[4] =  sb * camc; gq[5] =  sb * samc;   // U10 =  sb * e^{+i(a-c)/2}
    gq[6] =  cb * capc; gq[7] =  cb * sapc;   // U11 =  cb * e^{+i(a+c)/2}
    for (int j = 0; j < 8; ++j) gates[t * 8 + j] = gq[j];
  }
  __syncthreads();
  // Fused M = U19 (x) U20 (x) U21 (x) U22 over flat bits 3..0
  const int n = t >> 4, k = t & 15;
  float pr = 1.f, pi = 0.f;
  for (int qb = 19; qb <= 22; ++qb) {
    int bit = 22 - qb;                  // flat bit for this qubit
    int nb = (n >> bit) & 1;
    int kb = (k >> bit) & 1;
    const float* e = g + qb * 8 + (nb * 2 + kb) * 2;
    float er = e[0], ei = e[1];
    float nr = pr * er - pi * ei;
    float ni = pr * ei + pi * er;
    pr = nr; pi = ni;
  }
  mret[k * 16 + n] = pr;
  mimt[k * 16 + n] = pi;
  mngt[k * 16 + n] = -pi;
}

// ---------------------------------------------------------------------------
// K1: one pass handling flat bits 0..13 (qubits 9..22).
//  - load 2^14 amplitudes into LDS (re/im planes)
//  - 10 butterfly stages for bits 4..13
//  - WMMA 16x16 complex GEMM (4 chained K=4 f32 WMMAs per real GEMM) for bits 0..3
//  - store interleaved complex to workspace (b128 stores)
// ---------------------------------------------------------------------------
__global__ __launch_bounds__(256) void qc_low(
    const float* __restrict__ sre, const float* __restrict__ sim,
    const float* __restrict__ gates,
    const float* __restrict__ mret, const float* __restrict__ mimt,
    const float* __restrict__ mngt,
    float2* __restrict__ wstate)
{
  extern __shared__ float smem[];
  float* lre = smem;            // 16384 floats
  float* lim = smem + 16384;    // 16384 floats
  const int tid = threadIdx.x;
  const unsigned tileBase = (unsigned)blockIdx.x << 14;

  const float4* re4 = (const float4*)(sre + tileBase);
  const float4* im4 = (const float4*)(sim + tileBase);
  float4* lre4 = (float4*)lre;
  float4* lim4 = (float4*)lim;
  for (int i = tid; i < 4096; i += 256) {
    lre4[i] = re4[i];
    lim4[i] = im4[i];
  }
  __syncthreads();

  // Butterfly stages, flat bits 4..13
  for (int b = 4; b <= 13; ++b) {
    const float* gq = gates + (22 - b) * 8;
    float u00r = gq[0], u00i = gq[1], u01r = gq[2], u01i = gq[3];
    float u10r = gq[4], u10i = gq[5], u11r = gq[6], u11i = gq[7];
    for (int p = tid; p < 8192; p += 256) {
      int j0 = ((p >> b) << (b + 1)) | (p & ((1 << b) - 1));
      int j1 = j0 | (1 << b);
      float r0 = lre[j0], q0 = lim[j0];
      float r1 = lre[j1], q1 = lim[j1];
      lre[j0] = u00r * r0 - u00i * q0 + u01r * r1 - u01i * q1;
      lim[j0] = u00r * q0 + u00i * r0 + u01r * q1 + u01i * r1;
      lre[j1] = u10r * r0 - u10i * q0 + u11r * r1 - u11i * q1;
      lim[j1] = u10r * q0 + u10i * r0 + u11r * q1 + u11i * r1;
    }
    __syncthreads();
  }

  // WMMA stage: bits 0..3. State viewed as (1024 rows x 16 cols) per tile.
  // OutRe = InRe*MRe^T + InIm*(-MIm^T);  OutIm = InRe*MIm^T + InIm*MRe^T.
  const int lane = tid & 31;
  const int wave = tid >> 5;
  const int nn  = lane & 15;           // A: row M / B,D: col N
  const int khi = (lane >> 4) << 1;    // lanes 16..31 hold K=2,3 per 16x4 layout

  v2f Bre[4], Bim[4], Bng[4];
  for (int cc = 0; cc < 4; ++cc) {
    int k0 = 4 * cc + khi;
    Bre[cc].x = mret[k0 * 16 + nn]; Bre[cc].y = mret[(k0 + 1) * 16 + nn];
    Bim[cc].x = mimt[k0 * 16 + nn]; Bim[cc].y = mimt[(k0 + 1) * 16 + nn];
    Bng[cc].x = mngt[k0 * 16 + nn]; Bng[cc].y = mngt[(k0 + 1) * 16 + nn];
  }

  // Each wave owns rows [wave*128, wave*128+128): 8 tiles of 16 rows.
  // Constant trip count -> no exec-masked loop around the WMMAs.
  const int waveRowBase = wave << 7;
  for (int it = 0; it < 8; ++it) {
    const int rowBase = waveRowBase + (it << 4);
    const int arow = rowBase + nn;
    v2f Are[4], Aim[4];
    for (int cc = 0; cc < 4; ++cc) {
      int idx = arow * 16 + 4 * cc + khi;
      Are[cc].x = lre[idx]; Are[cc].y = lre[idx + 1];
      Aim[cc].x = lim[idx]; Aim[cc].y = lim[idx + 1];
    }
    v8f accR = {}; v8f accI = {};
    for (int cc = 0; cc < 4; ++cc)
      accR = __builtin_amdgcn_wmma_f32_16x16x4_f32(false, Are[cc], false, Bre[cc], (short)0, accR, false, false);
    for (int cc = 0; cc < 4; ++cc)
      accR = __builtin_amdgcn_wmma_f32_16x16x4_f32(false, Aim[cc], false, Bng[cc], (short)0, accR, false, false);
    for (int cc = 0; cc < 4; ++cc)
      accI = __builtin_amdgcn_wmma_f32_16x16x4_f32(false, Are[cc], false, Bim[cc], (short)0, accI, false, false);
    for (int cc = 0; cc < 4; ++cc)
      accI = __builtin_amdgcn_wmma_f32_16x16x4_f32(false, Aim[cc], false, Bre[cc], (short)0, accI, false, false);
    const int mbase = (lane >> 4) << 3;   // D: lanes 16..31 hold M=8..15
    for (int j = 0; j < 8; ++j) {
      int row = rowBase + mbase + j;
      lre[row * 16 + nn] = accR[j];
      lim[row * 16 + nn] = accI[j];
    }
  }
  __syncthreads();

  // Pack two adjacent complex values per thread -> b128 global stores.
  float4* w4 = (float4*)(wstate + tileBase);
  for (int i = tid; i < 8192; i += 256) {
    float4 v;
    v.x = lre[2 * i];     v.y = lim[2 * i];
    v.z = lre[2 * i + 1]; v.w = lim[2 * i + 1];
    w4[i] = v;
  }
}

// ---------------------------------------------------------------------------
// K2: one pass handling flat bits 14..22 (qubits 0..8), then the fused
// inverse-Gray scatter from the CNOT chain into planar re/im output.
// LDS layout [l*513 + h] (pitch 513 -> conflict-free for both access phases).
// ---------------------------------------------------------------------------
__global__ __launch_bounds__(256) void qc_high(
    const float2* __restrict__ wstate,
    const float* __restrict__ gates,
    float* __restrict__ outp)
{
  extern __shared__ float smem[];
  float* lre = smem;                 // 32 * 513
  float* lim = smem + 32 * 513;
  const int tid = threadIdx.x;
  const unsigned lowBase = (unsigned)blockIdx.x << 5;   // 32 low indices

  // b128 gather: two adjacent complex (adjacent low index, same high bits).
  for (int e = tid; e < 8192; e += 256) {
    int h  = e >> 4;
    int lp = (e & 15) << 1;
    const float4 v =
        *(const float4*)(wstate + (((unsigned)h << 14) | (lowBase + (unsigned)lp)));
    lre[lp * 513 + h]       = v.x;
    lim[lp * 513 + h]       = v.y;
    lre[(lp + 1) * 513 + h] = v.z;
    lim[(lp + 1) * 513 + h] = v.w;
  }
  __syncthreads();

  for (int b = 14; b <= 22; ++b) {
    const float* gq = gates + (22 - b) * 8;
    float u00r = gq[0], u00i = gq[1], u01r = gq[2], u01i = gq[3];
    float u10r = gq[4], u10i = gq[5], u11r = gq[6], u11i = gq[7];
    const int bb = b - 14;
    for (int p = tid; p < 8192; p += 256) {
      int l  = p >> 8;
      int hp = p & 255;
      int h0 = ((hp >> bb) << (bb + 1)) | (hp & ((1 << bb) - 1));
      int h1 = h0 | (1 << bb);
      int j0 = l * 513 + h0, j1 = l * 513 + h1;
      float r0 = lre[j0], q0 = lim[j0];
      float r1 = lre[j1], q1 = lim[j1];
      lre[j0] = u00r * r0 - u00i * q0 + u01r * r1 - u01i * q1;
      lim[j0] = u00r * q0 + u00i * r0 + u01r * q1 + u01i * r1;
      lre[j1] = u10r * r0 - u10i * q0 + u11r * r1 - u11i * q1;
      lim[j1] = u10r * q0 + u10i * r0 + u11r * q1 + u11i * r1;
    }
    __syncthreads();
  }

  // CNOT chain == Gray-code permutation: out[x] = s[x ^ (x>>1)].
  // We hold s[y]; write to x = invGray(y) (stays within each 32-block,
  // so stores remain cacheline-coalesced into the planar output).
  float* __restrict__ outRe = outp;
  float* __restrict__ outIm = outp + NSTATE;
  for (int e = tid; e < 16384; e += 256) {
    int h = e >> 5, l = e & 31;
    unsigned y = ((unsigned)h << 14) | (lowBase + (unsigned)l);
    unsigned x = y;
    x ^= x >> 1; x ^= x >> 2; x ^= x >> 4; x ^= x >> 8; x ^= x >> 16;
    outRe[x] = lre[l * 513 + h];
    outIm[x] = lim[l * 513 + h];
  }
}

// ---------------------------------------------------------------------------
extern "C" void kernel_launch(void* const* d_in, const int* in_sizes, int n_in,
                              void* d_out, int out_size, void* d_ws, size_t ws_size,
                              hipStream_t stream) {
  (void)in_sizes; (void)n_in; (void)out_size; (void)ws_size;
  const float* sre   = (const float*)d_in[0];
  const float* sim   = (const float*)d_in[1];
  const float* param = (const float*)d_in[2];

  // ws layout: [0, 2*NSTATE) floats = interleaved complex scratch (64 MB),
  // then gate tables (~3 KB).
  float*  ws     = (float*)d_ws;
  float2* wstate = (float2*)ws;
  float*  gates  = ws + 2ull * NSTATE;
  float*  mret   = gates + 23 * 8;
  float*  mimt   = mret + 256;
  float*  mngt   = mimt + 256;

  qc_prepare<<<1, 256, 0, stream>>>(param, gates, mret, mimt, mngt);
  qc_low<<<NSTATE / 16384, 256, 2 * 16384 * sizeof(float), stream>>>(
      sre, sim, gates, mret, mimt, mngt, wstate);
  qc_high<<<16384 / 32, 256, 2 * 32 * 513 * sizeof(float), stream>>>(
      wstate, gates, (float*)d_out);
}